// GCN_53919019434434
// MI455X (gfx1250) — compile-verified
//
#include <hip/hip_runtime.h>
#include <math.h>

// ---------------- problem constants (from reference) ----------------
#define NN   100000      // nodes (== 6250 * 16, exact M-tile fit)
#define EE   1600000     // edges
#define D_IN 128
#define D_H  64
#define D_OUT 40

typedef __attribute__((ext_vector_type(2))) float v2f;
typedef __attribute__((ext_vector_type(8))) float v8f;

static inline int cdiv_i(long long a, int b) { return (int)((a + b - 1) / b); }

// ---------------- degree pipeline ----------------
__global__ void k_init_deg(float* __restrict__ deg, int n) {
    int i = blockIdx.x * blockDim.x + threadIdx.x;
    if (i < n) deg[i] = 1.0f;                 // self-loop contributes 1
}

__global__ void k_deg_accum(const int* __restrict__ dst, float* __restrict__ deg, int e) {
    int i = blockIdx.x * blockDim.x + threadIdx.x;
    if (i < e) atomicAdd(&deg[dst[i]], 1.0f);
}

__global__ void k_inv_sqrt(float* __restrict__ deg, int n) {
    int i = blockIdx.x * blockDim.x + threadIdx.x;
    if (i < n) deg[i] = rsqrtf(deg[i]);       // deg buffer now holds deg^-1/2
}

__global__ void k_zero(float* __restrict__ p, int n) {
    int i = blockIdx.x * blockDim.x + threadIdx.x;
    if (i < n) p[i] = 0.0f;
}

// ---------------- weight prep: transpose + zero-pad columns ----------------
// WpT[c*K + k] = (c < NC) ? W[k*NC + c] : 0   for c in [0, NCP), NCP = 16*NT.
// Makes the per-lane B operand a contiguous 8-byte load with no guard.
__global__ void k_prep_wt(const float* __restrict__ W, float* __restrict__ WpT,
                          int K, int NC, int NCP) {
    int t = blockIdx.x * blockDim.x + threadIdx.x;     // t = c*K + k
    if (t >= NCP * K) return;
    int c = t / K;
    int k = t - c * K;
    WpT[t] = (c < NC) ? W[k * NC + c] : 0.0f;
}

// ---------------- WMMA f32 GEMM: Y[M,NC] = X[M,K] @ W[K,NC] ----------------
// One wave per 16x16 output tile. M must be a multiple of 16 (100000 = 6250*16).
// WpT is the transposed zero-padded weight panel [16*NT, K]; branchless inner loop.
template<int K, int NC, int NT>
__global__ void k_gemm_wmma(const float* __restrict__ X, const float* __restrict__ WpT,
                            float* __restrict__ Y, int mtiles) {
    int wave = (blockIdx.x * blockDim.x + threadIdx.x) >> 5;
    if (wave >= mtiles * NT) return;          // wave-uniform: EXEC stays all-1 for WMMA
    int mt = wave / NT;
    int nt = wave - mt * NT;

    int lane = threadIdx.x & 31;
    int half = lane >> 4;                     // 0: K = k0+0/1, 1: K = k0+2/3
    int l    = lane & 15;

    int rowA = mt * 16 + l;                   // A-matrix row for this lane
    int col  = nt * 16 + l;                   // B/D column for this lane
    const float* xrow = X   + (size_t)rowA * K + 2 * half;
    const float* brow = WpT + (size_t)col  * K + 2 * half;

    v8f c = {};
#pragma unroll
    for (int k0 = 0; k0 < K; k0 += 4) {
        v2f a = *(const v2f*)(xrow + k0);     // contiguous float2, row-major X
        v2f b = *(const v2f*)(brow + k0);     // contiguous float2, transposed W panel
        // D = A x B + C, full f32 precision, 16x16x4
        c = __builtin_amdgcn_wmma_f32_16x16x4_f32(
                /*neg_a=*/false, a, /*neg_b=*/false, b,
                /*c_mod=*/(short)0, c, /*reuse_a=*/false, /*reuse_b=*/false);
    }

    if (col < NC) {                           // one-time store guard (padded cols dropped)
        int rb = mt * 16 + 8 * half;          // C/D layout: VGPR v -> row v + 8*half
#pragma unroll
        for (int v = 0; v < 8; ++v)
            Y[(size_t)(rb + v) * NC + col] = c[v];
    }
}

// ---------------- edge scatter: agg[d] += inv[s]*inv[d] * h[s] ----------------
template<int F>
__global__ void k_edge_agg(const int* __restrict__ src, const int* __restrict__ dst,
                           const float* __restrict__ inv, const float* __restrict__ h,
                           float* __restrict__ agg, int e) {
    long long t = (long long)blockIdx.x * blockDim.x + threadIdx.x;
    if (t >= (long long)e * F) return;
    int edge = (int)(t / F);
    int j    = (int)(t - (long long)edge * F);
    int s = src[edge], d = dst[edge];
    float coef = inv[s] * inv[d];
    atomicAdd(&agg[(size_t)d * F + j], coef * h[(size_t)s * F + j]);
}

// ---------------- layer-1 combine: agg = relu(agg + (1/deg)*h + b) ----------------
__global__ void k_combine_relu(const float* __restrict__ h, const float* __restrict__ inv,
                               const float* __restrict__ b, float* __restrict__ agg, int n) {
    int t = blockIdx.x * blockDim.x + threadIdx.x;           // t = i*64 + j
    if (t >= n * D_H) return;
    int i = t >> 6;
    int j = t & 63;
    float selfc = inv[i] * inv[i];                           // = 1/deg[i]
    float v = agg[t] + selfc * h[t] + b[j];
    agg[t] = v > 0.0f ? v : 0.0f;
}

// ---------------- layer-2 combine + log_softmax over 40 classes ----------------
__global__ void k_combine_logsoftmax(const float* __restrict__ h2, const float* __restrict__ inv,
                                     const float* __restrict__ b2, const float* __restrict__ agg2,
                                     float* __restrict__ out, int n) {
    int i = blockIdx.x * blockDim.x + threadIdx.x;
    if (i >= n) return;
    float selfc = inv[i] * inv[i];
    float v[D_OUT];
    float m = -INFINITY;
#pragma unroll
    for (int j = 0; j < D_OUT; ++j) {
        float x = agg2[(size_t)i * D_OUT + j] + selfc * h2[(size_t)i * D_OUT + j] + b2[j];
        v[j] = x;
        m = fmaxf(m, x);
    }
    float s = 0.0f;
#pragma unroll
    for (int j = 0; j < D_OUT; ++j) s += expf(v[j] - m);
    float ls = logf(s) + m;
#pragma unroll
    for (int j = 0; j < D_OUT; ++j) out[(size_t)i * D_OUT + j] = v[j] - ls;
}

// ---------------- host-side orchestration ----------------
extern "C" void kernel_launch(void* const* d_in, const int* in_sizes, int n_in,
                              void* d_out, int out_size, void* d_ws, size_t ws_size,
                              hipStream_t stream) {
    (void)in_sizes; (void)n_in; (void)out_size; (void)ws_size;

    const float* feat = (const float*)d_in[0];   // [NN, 128]
    const int*   ei   = (const int*)  d_in[1];   // [2, EE]
    const float* W1   = (const float*)d_in[2];   // [128, 64]
    const float* b1   = (const float*)d_in[3];   // [64]
    const float* W2   = (const float*)d_in[4];   // [64, 40]
    const float* b2   = (const float*)d_in[5];   // [40]
    const int* src = ei;
    const int* dst = ei + EE;
    float* out = (float*)d_out;

    // workspace layout (~52 MB):
    //   deg | bufA (h1 -> h2) | bufB (agg1/x2 -> agg2) | W1^T padded | W2^T padded
    float* deg  = (float*)d_ws;
    float* bufA = deg  + ((NN + 255) & ~255);
    float* bufB = bufA + (size_t)NN * D_H;
    float* wt1  = bufB + (size_t)NN * D_H;       // [64  x 128]
    float* wt2  = wt1  + 64 * D_IN;              // [48  x  64]

    const int B = 256;

    // degree^-1/2
    k_init_deg <<<cdiv_i(NN, B), B, 0, stream>>>(deg, NN);
    k_deg_accum<<<cdiv_i(EE, B), B, 0, stream>>>(dst, deg, EE);
    k_inv_sqrt <<<cdiv_i(NN, B), B, 0, stream>>>(deg, NN);

    // weight panels (transpose + zero-pad): independent of degree pipeline
    k_prep_wt<<<cdiv_i(64 * D_IN, B), B, 0, stream>>>(W1, wt1, D_IN, D_H, 64);
    k_prep_wt<<<cdiv_i(48 * D_H,  B), B, 0, stream>>>(W2, wt2, D_H, D_OUT, 48);

    // ---- layer 1: h1 = feat @ W1 ; agg1 = scatter ; x2 = relu(agg1 + self + b1) ----
    {
        const int mtiles = NN / 16, NT = D_H / 16;           // 6250 x 4 tiles
        k_gemm_wmma<D_IN, D_H, NT>
            <<<cdiv_i((long long)mtiles * NT * 32, B), B, 0, stream>>>(feat, wt1, bufA, mtiles);
    }
    k_zero<<<cdiv_i((long long)NN * D_H, B), B, 0, stream>>>(bufB, NN * D_H);
    k_edge_agg<D_H><<<cdiv_i((long long)EE * D_H, B), B, 0, stream>>>(src, dst, deg, bufA, bufB, EE);
    k_combine_relu<<<cdiv_i((long long)NN * D_H, B), B, 0, stream>>>(bufA, deg, b1, bufB, NN);

    // ---- layer 2: h2 = x2 @ W2 ; agg2 = scatter ; out = log_softmax(...) ----
    {
        const int mtiles = NN / 16, NT = 3;                  // 48 cols padded, guard < 40
        k_gemm_wmma<D_H, D_OUT, NT>
            <<<cdiv_i((long long)mtiles * NT * 32, B), B, 0, stream>>>(bufB, wt2, bufA, mtiles);
    }
    k_zero<<<cdiv_i((long long)NN * D_OUT, B), B, 0, stream>>>(bufB, NN * D_OUT);
    k_edge_agg<D_OUT><<<cdiv_i((long long)EE * D_OUT, B), B, 0, stream>>>(src, dst, deg, bufA, bufB, EE);
    k_combine_logsoftmax<<<cdiv_i(NN, B), B, 0, stream>>>(bufA, deg, b2, bufB, out, NN);
}